// CGCNN_85186381349381
// MI455X (gfx1250) — compile-verified
//
#include <hip/hip_runtime.h>

#define BN_EPS 1e-5f

typedef float v2f __attribute__((ext_vector_type(2)));
typedef float v8f __attribute__((ext_vector_type(8)));

// ---------------------------------------------------------------------------
// FP32 GEMM via V_WMMA_F32_16X16X4_F32.
// One wave computes a 16 x (16*NT) C tile, K stepped by 4.
// A 16x4: lanes0-15 M=lane, vgpr0/1 = K0/K1; lanes16-31 = K2/K3 (half*2 offset).
// B 4x16 mirrors A with N in place of M.  C: vgpr r -> M = r + half*8, N = lane&15.
// Out-of-range rows/cols are CLAMPED for loads (always legal, unconditional)
// and guarded only at the store, so the inner loop is branch-free.
// ---------------------------------------------------------------------------
template <int NT>
__global__ void gemm_plain(const float* __restrict__ A, const float* __restrict__ Bm,
                           const float* __restrict__ bias, float* __restrict__ C,
                           int M, int K, int N) {
  int wid  = (blockIdx.x * blockDim.x + threadIdx.x) >> 5;  // wave-uniform
  int lane = threadIdx.x & 31;
  int nb     = (N + 16 * NT - 1) / (16 * NT);
  int mtiles = (M + 15) >> 4;
  if (wid >= mtiles * nb) return;
  int mt = wid / nb, nt = wid % nb;
  int half = lane >> 4, l16 = lane & 15;

  int arow = mt * 16 + l16; if (arow >= M) arow = M - 1;     // clamp
  const float* Ar = A + (size_t)arow * K + half * 2;

  int colb = nt * 16 * NT + l16;
  int cols[NT];
#pragma unroll
  for (int t = 0; t < NT; ++t) {
    int c = colb + 16 * t;
    cols[t] = c < N ? c : N - 1;                             // clamp
  }

  v8f zero8 = {0.f, 0.f, 0.f, 0.f, 0.f, 0.f, 0.f, 0.f};
  v8f acc[NT];
#pragma unroll
  for (int t = 0; t < NT; ++t) acc[t] = zero8;

#pragma unroll 4
  for (int k = 0; k < K; k += 4) {
    v2f a = *(const v2f*)(Ar + k);
    const float* B0 = Bm + (size_t)(k + half * 2) * N;
    const float* B1 = B0 + N;
#pragma unroll
    for (int t = 0; t < NT; ++t) {
      v2f b; b.x = B0[cols[t]]; b.y = B1[cols[t]];
      acc[t] = __builtin_amdgcn_wmma_f32_16x16x4_f32(false, a, false, b, (short)0,
                                                     acc[t], false, false);
    }
  }

  int rbase = mt * 16 + half * 8;
#pragma unroll
  for (int t = 0; t < NT; ++t) {
    int c = colb + 16 * t;
    if (c < N) {
      float bv = bias ? bias[c] : 0.f;
#pragma unroll
      for (int r = 0; r < 8; ++r) {
        int mr = rbase + r;
        if (mr < M) C[(size_t)mr * N + c] = acc[t][r] + bv;
      }
    }
  }
}

// Message GEMM with fused edge gather: A row = [ h[dst] | h[src] | e[row] ], K=192.
template <int NT>
__global__ void gemm_gather(const float* __restrict__ h, const float* __restrict__ e,
                            const int* __restrict__ srcI, const int* __restrict__ dstI,
                            const float* __restrict__ Bm, const float* __restrict__ bias,
                            float* __restrict__ C, int M, int N) {
  int wid  = (blockIdx.x * blockDim.x + threadIdx.x) >> 5;
  int lane = threadIdx.x & 31;
  int nb     = (N + 16 * NT - 1) / (16 * NT);
  int mtiles = (M + 15) >> 4;
  if (wid >= mtiles * nb) return;
  int mt = wid / nb, nt = wid % nb;
  int half = lane >> 4, l16 = lane & 15;

  int arow = mt * 16 + l16; if (arow >= M) arow = M - 1;
  // hoist gather indices out of the K loop
  const float* hd = h + (size_t)dstI[arow] * 64 + half * 2;
  const float* hs = h + (size_t)srcI[arow] * 64 + half * 2;
  const float* er = e + (size_t)arow * 64 + half * 2;

  int colb = nt * 16 * NT + l16;
  int cols[NT];
#pragma unroll
  for (int t = 0; t < NT; ++t) {
    int c = colb + 16 * t;
    cols[t] = c < N ? c : N - 1;
  }

  v8f zero8 = {0.f, 0.f, 0.f, 0.f, 0.f, 0.f, 0.f, 0.f};
  v8f acc[NT];
#pragma unroll
  for (int t = 0; t < NT; ++t) acc[t] = zero8;

  const float* segs[3] = {hd, hs, er};
#pragma unroll
  for (int s = 0; s < 3; ++s) {
    const float* Ar = segs[s];
    int koff = s * 64;
#pragma unroll 4
    for (int k = 0; k < 64; k += 4) {
      v2f a = *(const v2f*)(Ar + k);
      const float* B0 = Bm + (size_t)(koff + k + half * 2) * N;
      const float* B1 = B0 + N;
#pragma unroll
      for (int t = 0; t < NT; ++t) {
        v2f b; b.x = B0[cols[t]]; b.y = B1[cols[t]];
        acc[t] = __builtin_amdgcn_wmma_f32_16x16x4_f32(false, a, false, b, (short)0,
                                                       acc[t], false, false);
      }
    }
  }

  int rbase = mt * 16 + half * 8;
#pragma unroll
  for (int t = 0; t < NT; ++t) {
    int c = colb + 16 * t;
    if (c < N) {
      float bv = bias[c];
#pragma unroll
      for (int r = 0; r < 8; ++r) {
        int mr = rbase + r;
        if (mr < M) C[(size_t)mr * N + c] = acc[t][r] + bv;
      }
    }
  }
}

// ---------------------------------------------------------------------------
// Utility / pipeline kernels
// ---------------------------------------------------------------------------
__global__ void zero_kernel(float* __restrict__ p, size_t n) {
  size_t i  = (size_t)blockIdx.x * blockDim.x + threadIdx.x;
  size_t st = (size_t)gridDim.x * blockDim.x;
  for (; i < n; i += st) p[i] = 0.f;
}

__global__ void scale_rows_kernel(const float* __restrict__ x, const float* __restrict__ att,
                                  float* __restrict__ y, size_t n, int C) {
  size_t i = (size_t)blockIdx.x * blockDim.x + threadIdx.x;
  if (i < n) y[i] = x[i] * att[i % C];
}

// Column sums + sums-of-squares over X[R,C] (training-mode BatchNorm stats).
__global__ void col_stats_kernel(const float* __restrict__ X, float* __restrict__ sums,
                                 float* __restrict__ sqs, int R, int C) {
  int c = blockIdx.x * blockDim.x + threadIdx.x;
  if (c >= C) return;
  float s = 0.f, q = 0.f;
  for (int r = blockIdx.y; r < R; r += gridDim.y) {
    float v = X[(size_t)r * C + c];
    s += v; q += v * v;
  }
  atomicAdd(&sums[c], s);
  atomicAdd(&sqs[c], q);
}

// tot[E,128] -> BN -> split -> relu(filter)*relu(core) -> msg[E,64]
__global__ void bn_gate_kernel(const float* __restrict__ tot, const float* __restrict__ sums,
                               const float* __restrict__ sqs, const float* __restrict__ g,
                               const float* __restrict__ b, float* __restrict__ msg, int E) {
  size_t i = (size_t)blockIdx.x * blockDim.x + threadIdx.x;
  if (i >= (size_t)E * 64) return;
  size_t e = i >> 6; int c = (int)(i & 63);
  float invE = 1.f / (float)E;
  float m1 = sums[c] * invE,      v1 = sqs[c] * invE - m1 * m1;
  float m2 = sums[c + 64] * invE, v2 = sqs[c + 64] * invE - m2 * m2;
  float nf = (tot[e * 128 + c]      - m1) * rsqrtf(v1 + BN_EPS) * g[c]      + b[c];
  float nc = (tot[e * 128 + c + 64] - m2) * rsqrtf(v2 + BN_EPS) * g[c + 64] + b[c + 64];
  msg[i] = fmaxf(nf, 0.f) * fmaxf(nc, 0.f);
}

__global__ void scatter_add_kernel(const float* __restrict__ msg, const int* __restrict__ dst,
                                   float* __restrict__ aggr, int E) {
  size_t i = (size_t)blockIdx.x * blockDim.x + threadIdx.x;
  if (i >= (size_t)E * 64) return;
  size_t e = i >> 6; int c = (int)(i & 63);
  atomicAdd(&aggr[(size_t)dst[e] * 64 + c], msg[i]);
}

// h_out = relu(h + BN(aggr))
__global__ void bn2_update_kernel(const float* __restrict__ h, const float* __restrict__ aggr,
                                  const float* __restrict__ sums, const float* __restrict__ sqs,
                                  const float* __restrict__ g, const float* __restrict__ b,
                                  float* __restrict__ hout, int Nn) {
  size_t i = (size_t)blockIdx.x * blockDim.x + threadIdx.x;
  if (i >= (size_t)Nn * 64) return;
  int c = (int)(i & 63);
  float invN = 1.f / (float)Nn;
  float m = sums[c] * invN, v = sqs[c] * invN - m * m;
  float val = h[i] + (aggr[i] - m) * rsqrtf(v + BN_EPS) * g[c] + b[c];
  hout[i] = fmaxf(val, 0.f);
}

// In-place BN + LeakyReLU(0.01) over X[R,C]
__global__ void bn_lrelu_kernel(float* __restrict__ X, const float* __restrict__ sums,
                                const float* __restrict__ sqs, const float* __restrict__ g,
                                const float* __restrict__ b, int R, int C) {
  size_t i = (size_t)blockIdx.x * blockDim.x + threadIdx.x;
  if (i >= (size_t)R * C) return;
  int c = (int)(i % C);
  float invR = 1.f / (float)R;
  float m = sums[c] * invR, var = sqs[c] * invR - m * m;
  float v = (X[i] - m) * rsqrtf(var + BN_EPS) * g[c] + b[c];
  X[i] = v > 0.f ? v : 0.01f * v;
}

// AutoDiscretization: edge scalar -> softmax over 100 bins -> @lut -> e[E,64]
__global__ void autodisc_kernel(const float* __restrict__ edge_attr,
                                const float* __restrict__ w1, const float* __restrict__ b1,
                                const float* __restrict__ w2, const float* __restrict__ b2,
                                const float* __restrict__ alphap, const float* __restrict__ lut,
                                float* __restrict__ eout, int E) {
  __shared__ float v2s[100];
  __shared__ float ps[100];
  __shared__ float red[4];
  int e = blockIdx.x;
  if (e >= E) return;
  int t = threadIdx.x;  // 128 threads
  float ea = edge_attr[e];
  float alpha = alphap[0];
  if (t < 100) {
    float v = ea * w1[t] + b1[t];
    v2s[t] = v > 0.f ? v : 0.01f * v;   // leaky_relu
  }
  __syncthreads();
  float v3 = -3.0e38f;
  if (t < 100) {
    float s = b2[t] + alpha * v2s[t];
    for (int k = 0; k < 100; ++k) s += v2s[k] * w2[k * 100 + t];
    v3 = s;
  }
  float m = v3;
  for (int o = 16; o; o >>= 1) m = fmaxf(m, __shfl_xor(m, o, 32));
  if ((t & 31) == 0) red[t >> 5] = m;
  __syncthreads();
  m = fmaxf(fmaxf(red[0], red[1]), fmaxf(red[2], red[3]));
  __syncthreads();
  float p = (t < 100) ? __expf(v3 - m) : 0.f;
  float s = p;
  for (int o = 16; o; o >>= 1) s += __shfl_xor(s, o, 32);
  if ((t & 31) == 0) red[t >> 5] = s;
  __syncthreads();
  s = red[0] + red[1] + red[2] + red[3];
  if (t < 100) ps[t] = p / s;
  __syncthreads();
  if (t < 64) {
    float acc = 0.f;
    for (int k = 0; k < 100; ++k) acc += ps[k] * lut[k * 64 + t];
    eout[(size_t)e * 64 + t] = acc;
  }
}

// Row softmax over 64 columns, one wave per row (lane handles 2 cols).
__global__ void softmax64_kernel(float* __restrict__ X, long long R) {
  int wave = threadIdx.x >> 5, lane = threadIdx.x & 31;
  long long row = (long long)blockIdx.x * 8 + wave;
  if (row >= R) return;
  float* p = X + (size_t)row * 64;
  float a = p[lane], b = p[lane + 32];
  float m = fmaxf(a, b);
  for (int o = 16; o; o >>= 1) m = fmaxf(m, __shfl_xor(m, o, 32));
  float ea = __expf(a - m), eb = __expf(b - m);
  float s = ea + eb;
  for (int o = 16; o; o >>= 1) s += __shfl_xor(s, o, 32);
  float inv = 1.f / s;
  p[lane] = ea * inv; p[lane + 32] = eb * inv;
}

// crys[b,c] = sum over 4 levels x 5 atoms of pp[(l*N + b*5 + a), c]
__global__ void pool_kernel(const float* __restrict__ pp, float* __restrict__ crys,
                            int B, int Nn) {
  int i = blockIdx.x * blockDim.x + threadIdx.x;
  if (i >= B * 64) return;
  int b = i >> 6, c = i & 63;
  float s = 0.f;
  for (int l = 0; l < 4; ++l)
    for (int a = 0; a < 5; ++a)
      s += pp[((size_t)l * Nn + (size_t)b * 5 + a) * 64 + c];
  crys[i] = s;
}

// Wf[k,j] = sum_r W[r*64+k, j]  (folds tiled-input decoder weights)
__global__ void fold_kernel(const float* __restrict__ W, float* __restrict__ Wf,
                            int Nout, int reps) {
  int i = blockIdx.x * blockDim.x + threadIdx.x;
  if (i >= 64 * Nout) return;
  int k = i / Nout, j = i % Nout;
  float s = 0.f;
  for (int r = 0; r < reps; ++r) s += W[((size_t)r * 64 + k) * Nout + j];
  Wf[(size_t)k * Nout + j] = s;
}

__global__ void dp_out_kernel(const float* __restrict__ hp, const float* __restrict__ Wo,
                              const float* __restrict__ bo, float* __restrict__ out, int B) {
  int b = blockIdx.x * blockDim.x + threadIdx.x;
  if (b >= B) return;
  float s = bo[0];
  for (int k = 0; k < 32; ++k) s += hp[(size_t)b * 32 + k] * Wo[k];
  out[b] = s;
}

// In-place log-softmax over rows of 86, one wave per row (lane covers 3 cols).
__global__ void logsoftmax86_kernel(float* __restrict__ X, int R) {
  int wave = threadIdx.x >> 5, lane = threadIdx.x & 31;
  long long row = (long long)blockIdx.x * 8 + wave;
  if (row >= R) return;
  float* p = X + (size_t)row * 86;
  float x0 = p[lane], x1 = p[lane + 32];
  bool has2 = lane < 22;                      // 86 = 64 + 22
  float x2 = has2 ? p[lane + 64] : -3.0e38f;
  float m = fmaxf(fmaxf(x0, x1), x2);
  for (int o = 16; o; o >>= 1) m = fmaxf(m, __shfl_xor(m, o, 32));
  float s = __expf(x0 - m) + __expf(x1 - m) + (has2 ? __expf(x2 - m) : 0.f);
  for (int o = 16; o; o >>= 1) s += __shfl_xor(s, o, 32);
  float l = m + __logf(s);
  p[lane] = x0 - l; p[lane + 32] = x1 - l;
  if (has2) p[lane + 64] = x2 - l;
}

// ---------------------------------------------------------------------------
// Host orchestration
// ---------------------------------------------------------------------------
extern "C" void kernel_launch(void* const* d_in, const int* in_sizes, int n_in,
                              void* d_out, int out_size, void* d_ws, size_t ws_size,
                              hipStream_t stream) {
  (void)n_in; (void)out_size; (void)ws_size;
  const float* x         = (const float*)d_in[0];
  const int*   edge_idx  = (const int*)d_in[1];
  const float* edge_attr = (const float*)d_in[2];
  const float* attention = (const float*)d_in[4];
  const float* emb_W     = (const float*)d_in[5];
  const float* emb_b     = (const float*)d_in[6];
  const float* ad_w1     = (const float*)d_in[7];
  const float* ad_b1     = (const float*)d_in[8];
  const float* ad_w2     = (const float*)d_in[9];
  const float* ad_b2     = (const float*)d_in[10];
  const float* ad_alpha  = (const float*)d_in[11];
  const float* ad_lut    = (const float*)d_in[12];
  const float* fc_W      = (const float*)d_in[13];
  const float* fc_b      = (const float*)d_in[14];
  const float* bn1_g     = (const float*)d_in[15];
  const float* bn1_b     = (const float*)d_in[16];
  const float* bn2_g     = (const float*)d_in[17];
  const float* bn2_b     = (const float*)d_in[18];
  const float* pp_W      = (const float*)d_in[19];
  const float* pp_b      = (const float*)d_in[20];
  const float* dp_W1     = (const float*)d_in[21];
  const float* dp_b1     = (const float*)d_in[22];
  const float* dp_g      = (const float*)d_in[23];
  const float* dp_beta   = (const float*)d_in[24];
  const float* dp_Wo     = (const float*)d_in[25];
  const float* dp_bo     = (const float*)d_in[26];
  const float* de_Wo     = (const float*)d_in[27];
  const float* de_bo     = (const float*)d_in[28];
  const float* dc_W1     = (const float*)d_in[29];
  const float* dc_b1     = (const float*)d_in[30];
  const float* dc_g      = (const float*)d_in[31];
  const float* dc_beta   = (const float*)d_in[32];
  const float* dc_Wo     = (const float*)d_in[33];
  const float* dc_bo     = (const float*)d_in[34];

  const int Nn = in_sizes[0] / 92;   // 81920 nodes
  const int E  = in_sizes[2];        // 600000 edges
  const int B  = Nn / 5;             // 16384 crystals
  const int* srcI = edge_idx;        // edge_index[0]
  const int* dstI = edge_idx + E;    // edge_index[1]

  // ---- workspace layout (floats) ----
  float* ws = (float*)d_ws;
  float* feats = ws;                               // 4 * N * 64
  float* e_msg = feats + (size_t)4 * Nn * 64;      // E * 64 (edge feat, then msg)
  float* tot   = e_msg + (size_t)E * 64;           // E * 128
  float* aggr  = tot + (size_t)E * 128;            // N * 64
  float* stats = aggr + (size_t)Nn * 64;           // 1024 (sums | sqs @ +512)
  float* sums = stats, *sqs = stats + 512;
  // aliases inside the tot region (tot is only live inside each conv iteration)
  float* xs = tot;                                 // N*92 pre-scaled x (pre-conv)
  float* pp_buf = tot;                             // 4N * 64 (post-conv)
  float* crys = pp_buf + (size_t)4 * Nn * 64;      // B * 64
  float* hp   = crys + (size_t)B * 64;             // B * 32
  float* hc   = hp + (size_t)B * 32;               // B * 352
  float* W5   = hc + (size_t)B * 352;              // 64 * 430
  float* W11  = W5 + (size_t)64 * 430;             // 64 * 352

  float* out_prop   = (float*)d_out;               // [B]
  float* out_coords = out_prop + B;                // [B,11]
  float* out_elem   = out_coords + (size_t)B * 11; // [B*5,86]

  auto gemm = [&](const float* A, const float* Bm, const float* bias, float* C,
                  int M, int K, int N, int NT) {
    int nb = (N + 16 * NT - 1) / (16 * NT);
    int blocks = ((((M + 15) / 16) * nb) + 7) / 8;
    switch (NT) {
      case 1: gemm_plain<1><<<blocks, 256, 0, stream>>>(A, Bm, bias, C, M, K, N); break;
      case 2: gemm_plain<2><<<blocks, 256, 0, stream>>>(A, Bm, bias, C, M, K, N); break;
      case 4: gemm_plain<4><<<blocks, 256, 0, stream>>>(A, Bm, bias, C, M, K, N); break;
      default: gemm_plain<8><<<blocks, 256, 0, stream>>>(A, Bm, bias, C, M, K, N); break;
    }
  };
  auto zero = [&](float* p, size_t n) {
    int blocks = (int)(((n + 255) / 256 < 8192) ? (n + 255) / 256 : 8192);
    zero_kernel<<<blocks, 256, 0, stream>>>(p, n);
  };
  auto stats_of = [&](const float* X, int R, int C) {
    zero(stats, 1024);
    dim3 g((C + 255) / 256, 256);
    col_stats_kernel<<<g, 256, 0, stream>>>(X, sums, sqs, R, C);
  };

  // 1) embedding: h0 = (x*attention) @ emb_W + emb_b
  {
    size_t n = (size_t)Nn * 92;
    scale_rows_kernel<<<(int)((n + 255) / 256), 256, 0, stream>>>(x, attention, xs, n, 92);
    gemm(xs, emb_W, emb_b, feats, Nn, 92, 64, 4);
  }

  // 2) conv layers
  for (int i = 0; i < 3; ++i) {
    const float* h = feats + (size_t)i * Nn * 64;
    float* hnext   = feats + (size_t)(i + 1) * Nn * 64;
    // edge feature discretization -> e_msg[E,64]
    autodisc_kernel<<<E, 128, 0, stream>>>(edge_attr, ad_w1 + (size_t)i * 100,
                                           ad_b1 + (size_t)i * 100, ad_w2 + (size_t)i * 10000,
                                           ad_b2 + (size_t)i * 100, ad_alpha + i,
                                           ad_lut + (size_t)i * 6400, e_msg, E);
    // message GEMM (fused gather): tot[E,128], wave computes 16x128
    {
      int blocks = (((E + 15) / 16) + 7) / 8;   // nb==1 at NT=8, N=128
      gemm_gather<8><<<blocks, 256, 0, stream>>>(h, e_msg, srcI, dstI,
                                                 fc_W + (size_t)i * 192 * 128,
                                                 fc_b + (size_t)i * 128, tot, E, 128);
    }
    // BN1 + gated product -> msg (overwrites e_msg)
    stats_of(tot, E, 128);
    {
      size_t n = (size_t)E * 64;
      bn_gate_kernel<<<(int)((n + 255) / 256), 256, 0, stream>>>(
          tot, sums, sqs, bn1_g + (size_t)i * 128, bn1_b + (size_t)i * 128, e_msg, E);
    }
    // scatter-add into aggr
    zero(aggr, (size_t)Nn * 64);
    {
      size_t n = (size_t)E * 64;
      scatter_add_kernel<<<(int)((n + 255) / 256), 256, 0, stream>>>(e_msg, dstI, aggr, E);
    }
    // BN2 + residual relu -> hnext
    stats_of(aggr, Nn, 64);
    {
      size_t n = (size_t)Nn * 64;
      bn2_update_kernel<<<(int)((n + 255) / 256), 256, 0, stream>>>(
          h, aggr, sums, sqs, bn2_g + (size_t)i * 64, bn2_b + (size_t)i * 64, hnext, Nn);
    }
  }

  // 3) pooling: softmax(feats @ pp_W + pp_b) summed over levels & atoms
  gemm(feats, pp_W, pp_b, pp_buf, 4 * Nn, 64, 64, 4);
  {
    long long R = (long long)4 * Nn;
    softmax64_kernel<<<(int)((R + 7) / 8), 256, 0, stream>>>(pp_buf, R);
    pool_kernel<<<(B * 64 + 255) / 256, 256, 0, stream>>>(pp_buf, crys, B, Nn);
  }

  // 4) decoder_prop: BN(crys@W1+b1) -> leaky -> @Wo+bo
  gemm(crys, dp_W1, dp_b1, hp, B, 64, 32, 2);
  stats_of(hp, B, 32);
  bn_lrelu_kernel<<<((size_t)B * 32 + 255) / 256, 256, 0, stream>>>(hp, sums, sqs,
                                                                    dp_g, dp_beta, B, 32);
  dp_out_kernel<<<(B + 255) / 256, 256, 0, stream>>>(hp, dp_Wo, dp_bo, out_prop, B);

  // 5) decoder_element: fold tile(.,5) weights, GEMM into d_out, log_softmax rows of 86
  fold_kernel<<<(64 * 430 + 255) / 256, 256, 0, stream>>>(de_Wo, W5, 430, 5);
  gemm(crys, W5, de_bo, out_elem, B, 64, 430, 4);
  {
    int R = B * 5;
    logsoftmax86_kernel<<<(R + 7) / 8, 256, 0, stream>>>(out_elem, R);
  }

  // 6) decoder_coords: fold tile(.,11) weights, GEMM, BN+leaky, final GEMM [B,11]
  fold_kernel<<<(64 * 352 + 255) / 256, 256, 0, stream>>>(dc_W1, W11, 352, 11);
  gemm(crys, W11, dc_b1, hc, B, 64, 352, 4);
  stats_of(hc, B, 352);
  bn_lrelu_kernel<<<((size_t)B * 352 + 255) / 256, 256, 0, stream>>>(hc, sums, sqs,
                                                                     dc_g, dc_beta, B, 352);
  gemm(hc, dc_Wo, dc_bo, out_coords, B, 352, 11, 1);
}